// LearnableMidiSynth_1434519077429
// MI455X (gfx1250) — compile-verified
//
#include <hip/hip_runtime.h>
#include <math.h>

typedef __attribute__((ext_vector_type(2))) float v2f;
typedef __attribute__((ext_vector_type(8))) float v8f;

#define NNOTE   256
#define HID     32
#define NHARM   8
#define ZL      2
#define WIN     1024
#define MAXLEN  32768
#define TOUT    4194304
#define CHUNK   4096

__device__ __forceinline__ float sigmoid_f(float x) {
    return 1.0f / (1.0f + __expf(-x));
}

// ---------------------------------------------------------------------------
// Kernel A: per-note parameter MLPs.
// One block, 256 threads (8 wave32s). Hidden layers (K=32 GEMMs) run on
// V_WMMA_F32_16X16X4_F32; each wave owns two 16-note M-tiles (EXEC all-1s,
// no divergence around WMMA). Activations & zero-padded weights staged in LDS
// using the ISA 16x4 A-layout / 16x16 C-layout.
// ---------------------------------------------------------------------------
__global__ void __launch_bounds__(256) midi_params_kernel(
    const float* __restrict__ freq, const float* __restrict__ velocity,
    const int*   __restrict__ starts,
    const float* __restrict__ te_w1, const float* __restrict__ te_b1,
    const float* __restrict__ te_w2, const float* __restrict__ te_b2,
    const float* __restrict__ hs_w1, const float* __restrict__ hs_b1,
    const float* __restrict__ hs_w2, const float* __restrict__ hs_b2,
    float* __restrict__ hamps_out, float* __restrict__ vel_out)
{
    __shared__ float sH[NNOTE * HID];  // 32 KB activations (A matrix, row-major)
    __shared__ float sO[NNOTE * 16];   // 16 KB padded GEMM output
    __shared__ float sW[HID * 16];     //  2 KB padded B matrix (32 x 16)
    __shared__ float sB[16];           // padded bias

    const int t    = threadIdx.x;        // == note id for elementwise phases
    const int lane = t & 31;
    const int wave = t >> 5;
    const int col  = lane & 15;
    const int hi   = (lane >= 16) ? 1 : 0;
    const int kb   = hi ? 2 : 0;         // A/B K sub-offset per half-wave

    // ---- time embedder layer 1: relu(tnorm * w1 + b1), K=1 outer product ----
    {
        float tn = (float)starts[t] / (float)TOUT;
        #pragma unroll
        for (int j = 0; j < HID; ++j)
            sH[t * HID + j] = fmaxf(fmaf(tn, te_w1[j], te_b1[j]), 0.0f);
    }
    // pad te_w2 (32x2) -> sW (32x16 zero-padded), pad bias
    for (int i = t; i < HID * 16; i += 256) {
        int k = i >> 4, j = i & 15;
        sW[i] = (j < ZL) ? te_w2[k * ZL + j] : 0.0f;
    }
    if (t < 16) sB[t] = (t < ZL) ? te_b2[t] : 0.0f;
    __syncthreads();

    // ---- time embedder layer 2: [256x32]@[32x16pad] via WMMA f32 16x16x4 ----
    #pragma unroll
    for (int ti = 0; ti < 2; ++ti) {
        const int tile = wave * 2 + ti;
        const int m    = tile * 16 + col;
        float bn = sB[col];
        v8f c;
        #pragma unroll
        for (int r = 0; r < 8; ++r) c[r] = bn;   // bias broadcast per column
        #pragma unroll
        for (int kc = 0; kc < 8; ++kc) {
            const int k0 = kc * 4 + kb;
            v2f a, b;
            a[0] = sH[m * HID + k0];
            a[1] = sH[m * HID + k0 + 1];
            b[0] = sW[k0 * 16 + col];
            b[1] = sW[(k0 + 1) * 16 + col];
            c = __builtin_amdgcn_wmma_f32_16x16x4_f32(
                    false, a, false, b, (short)0, c, false, false);
        }
        #pragma unroll
        for (int r = 0; r < 8; ++r) {
            const int row = tile * 16 + r + (hi ? 8 : 0);
            sO[row * 16 + col] = c[r];
        }
    }
    __syncthreads();

    // ---- harmonic scaler layer 1: x=[freq, lat0, lat1], K=3 ----
    {
        float f0 = freq[t];
        float l0 = sO[t * 16 + 0];
        float l1 = sO[t * 16 + 1];
        #pragma unroll
        for (int j = 0; j < HID; ++j) {
            float h = fmaf(f0, hs_w1[j],
                      fmaf(l0, hs_w1[HID + j],
                      fmaf(l1, hs_w1[2 * HID + j], hs_b1[j])));
            sH[t * HID + j] = fmaxf(h, 0.0f);
        }
    }
    // pad hs_w2 (32x9) -> sW, pad bias
    for (int i = t; i < HID * 16; i += 256) {
        int k = i >> 4, j = i & 15;
        sW[i] = (j < NHARM + 1) ? hs_w2[k * (NHARM + 1) + j] : 0.0f;
    }
    if (t < 16) sB[t] = (t < NHARM + 1) ? hs_b2[t] : 0.0f;
    __syncthreads();

    // ---- harmonic scaler layer 2: [256x32]@[32x16pad] via WMMA ----
    #pragma unroll
    for (int ti = 0; ti < 2; ++ti) {
        const int tile = wave * 2 + ti;
        const int m    = tile * 16 + col;
        float bn = sB[col];
        v8f c;
        #pragma unroll
        for (int r = 0; r < 8; ++r) c[r] = bn;
        #pragma unroll
        for (int kc = 0; kc < 8; ++kc) {
            const int k0 = kc * 4 + kb;
            v2f a, b;
            a[0] = sH[m * HID + k0];
            a[1] = sH[m * HID + k0 + 1];
            b[0] = sW[k0 * 16 + col];
            b[1] = sW[(k0 + 1) * 16 + col];
            c = __builtin_amdgcn_wmma_f32_16x16x4_f32(
                    false, a, false, b, (short)0, c, false, false);
        }
        #pragma unroll
        for (int r = 0; r < 8; ++r) {
            const int row = tile * 16 + r + (hi ? 8 : 0);
            sO[row * 16 + col] = c[r];
        }
    }
    __syncthreads();

    // ---- readout: sigmoids -> hamps[n][8], vel[n] ----
    {
        float vg = sigmoid_f(sO[t * 16 + NHARM]);
        vel_out[t] = velocity[t] * (1.0f / 127.0f) * vg;
        #pragma unroll
        for (int j = 0; j < NHARM; ++j)
            hamps_out[t * NHARM + j] = sigmoid_f(sO[t * 16 + j]);
    }
}

// ---------------------------------------------------------------------------
// Kernel B: additive synthesis + overlap-add scatter.
// grid = (256 notes, 8 chunks of 4096 samples), block = 256 threads.
// One sincos per sample; harmonics 2..8 via Chebyshev recurrence
// sin((k+1)t) = 2cos(t) sin(kt) - sin((k-1)t). Hann fade via one fast cos.
// ---------------------------------------------------------------------------
__global__ void __launch_bounds__(256) midi_synth_kernel(
    const float* __restrict__ freq, const int* __restrict__ starts,
    const int*   __restrict__ lengths,
    const float* __restrict__ hamps, const float* __restrict__ vel,
    float* __restrict__ out)
{
    const int note = blockIdx.x;
    const int base = blockIdx.y * CHUNK;
    const int t    = threadIdx.x;

    __shared__ float sh[NHARM];
    __shared__ float sfv[2];   // freq, vel
    __shared__ int   ssl[2];   // start, length
    if (t < NHARM) sh[t] = hamps[note * NHARM + t];
    if (t == 8)  sfv[0] = freq[note];
    if (t == 9)  sfv[1] = vel[note];
    if (t == 10) ssl[0] = starts[note];
    if (t == 11) ssl[1] = lengths[note];
    __syncthreads();

    const int len = ssl[1];
    if (base >= len) return;               // whole chunk beyond note end
    const int   start = ssl[0];
    const float f     = sfv[0];
    const float v     = sfv[1];
    float hh[NHARM];
    #pragma unroll
    for (int k = 0; k < NHARM; ++k) hh[k] = sh[k];

    const float hann_k = 6.28318530717958647692f / (float)WIN;

    #pragma unroll 4
    for (int i = t; i < CHUNK; i += 256) {
        const int l = base + i;
        if (l >= len) continue;            // seg masked to zero beyond length
        float th = f * (float)l;
        float s1, c1;
        __sincosf(th, &s1, &c1);
        const float twoc = 2.0f * c1;
        // Chebyshev: 8 harmonics from one sincos
        float sprev = 0.0f, scur = s1, acc = 0.0f;
        #pragma unroll
        for (int k = 0; k < NHARM; ++k) {
            acc = fmaf(scur, hh[k], acc);
            float snext = fmaf(twoc, scur, -sprev);
            sprev = scur;
            scur  = snext;
        }
        // periodic-Hann fade over the final WIN samples
        const int wpos = l - (len - WIN);
        float factor = 1.0f;
        if (wpos >= 0)
            factor = 0.5f * (1.0f - __cosf((float)wpos * hann_k));
        atomicAdd(&out[start + l], acc * v * factor);
    }
}

extern "C" void kernel_launch(void* const* d_in, const int* in_sizes, int n_in,
                              void* d_out, int out_size, void* d_ws, size_t ws_size,
                              hipStream_t stream) {
    const float* freq     = (const float*)d_in[0];
    const float* velocity = (const float*)d_in[1];
    const int*   starts   = (const int*)  d_in[2];
    const int*   lengths  = (const int*)  d_in[3];
    const float* outinit  = (const float*)d_in[4];
    const float* te_w1    = (const float*)d_in[5];
    const float* te_b1    = (const float*)d_in[6];
    const float* te_w2    = (const float*)d_in[7];
    const float* te_b2    = (const float*)d_in[8];
    const float* hs_w1    = (const float*)d_in[9];
    const float* hs_b1    = (const float*)d_in[10];
    const float* hs_w2    = (const float*)d_in[11];
    const float* hs_b2    = (const float*)d_in[12];
    float* out = (float*)d_out;

    float* hamps = (float*)d_ws;                 // 256*8 floats
    float* vel   = hamps + NNOTE * NHARM;        // 256 floats

    // faithful init: result = output.at[flat].add(seg), starting from `output`
    hipMemcpyAsync(d_out, outinit, (size_t)out_size * sizeof(float),
                   hipMemcpyDeviceToDevice, stream);

    midi_params_kernel<<<1, 256, 0, stream>>>(
        freq, velocity, starts,
        te_w1, te_b1, te_w2, te_b2,
        hs_w1, hs_b1, hs_w2, hs_b2,
        hamps, vel);

    dim3 grid(NNOTE, MAXLEN / CHUNK);
    midi_synth_kernel<<<grid, 256, 0, stream>>>(
        freq, starts, lengths, hamps, vel, out);
}